// GAT_37761352466884
// MI455X (gfx1250) — compile-verified
//
#include <hip/hip_runtime.h>
#include <hip/hip_bf16.h>
#include <math.h>

// Problem constants (match reference)
#define F_IN 13
#define EDIM 2
#define HC1  64   // H1*C1
#define C2C  8
#define GC   64

typedef float v2f __attribute__((ext_vector_type(2)));
typedef float v8f __attribute__((ext_vector_type(8)));

static __device__ __forceinline__ void atomicMaxF(float* addr, float val) {
    // float atomic max via int/uint monotone punning (works incl. -inf init)
    if (val >= 0.0f) atomicMax((int*)addr, __float_as_int(val));
    else             atomicMin((unsigned int*)addr, (unsigned int)__float_as_int(val));
}

__global__ void fill_f32(float* __restrict__ p, float v, int n) {
    int i = blockIdx.x * blockDim.x + threadIdx.x;
    if (i < n) p[i] = v;
}

// ---- self loops: mean incoming edge_attr per node ----
__global__ void loop_sum_kernel(const int* __restrict__ ei, const float* __restrict__ ea,
                                float* __restrict__ lsum, float* __restrict__ lcnt, int E) {
    int e = blockIdx.x * blockDim.x + threadIdx.x;
    if (e >= E) return;
    int d = ei[E + e];
    atomicAdd(&lsum[2 * d + 0], ea[2 * e + 0]);
    atomicAdd(&lsum[2 * d + 1], ea[2 * e + 1]);
    atomicAdd(&lcnt[d], 1.0f);
}
__global__ void loop_div_kernel(float* __restrict__ lsum, const float* __restrict__ lcnt, int N) {
    int n = blockIdx.x * blockDim.x + threadIdx.x;
    if (n >= N) return;
    float c = fmaxf(lcnt[n], 1.0f);
    lsum[2 * n + 0] /= c;
    lsum[2 * n + 1] /= c;
}

// ---- tiny precompute: v_edge[d,h] = sum_c We[d, h*C+c] * a_edge[h,c] ----
__global__ void vedge_kernel(const float* __restrict__ We1, const float* __restrict__ ae1,
                             const float* __restrict__ We2, const float* __restrict__ ae2,
                             float* __restrict__ ve1, float* __restrict__ ve2) {
    int t = threadIdx.x;
    if (t < 16) {                     // layer1: d=t/8, h=t%8
        int d = t >> 3, h = t & 7;
        float s = 0.f;
        for (int c = 0; c < 8; ++c) s += We1[d * 64 + h * 8 + c] * ae1[h * 8 + c];
        ve1[t] = s;                   // ve1[d*8+h]
    } else if (t < 18) {              // layer2: H2=1
        int d = t - 16;
        float s = 0.f;
        for (int c = 0; c < 8; ++c) s += We2[d * 8 + c] * ae2[c];
        ve2[d] = s;
    }
}

// ---- WMMA fp32 GEMM: xl1[N,64] = x[N,13] @ W1[13,64] (K padded to 16) ----
// One wave per 16-row tile; A and B staged through LDS with branch-free
// clamped loads so WMMA operands come from ds_load_b64 / ds_load_b32.
__global__ __launch_bounds__(32) void gemm1_wmma(const float* __restrict__ x,
                                                 const float* __restrict__ W1,
                                                 float* __restrict__ xl1, int N) {
    __shared__ float sA[16][16];   // x tile, K padded 13 -> 16
    __shared__ float sB[16][64];   // W1 rows padded 13 -> 16
    const int row0 = blockIdx.x * 16;
    const int t = threadIdx.x;

    // stage A tile: 256 elems, 8 per lane (clamp + select, no branches)
    #pragma unroll
    for (int i = 0; i < 8; ++i) {
        int idx = i * 32 + t;
        int r = idx >> 4, c = idx & 15;
        int rr = min(row0 + r, N - 1);
        int cc = min(c, F_IN - 1);
        float v = x[(size_t)rr * F_IN + cc];
        sA[r][c] = (c < F_IN) ? v : 0.0f;
    }
    // stage B (all of W1): 1024 elems, 32 per lane
    #pragma unroll
    for (int i = 0; i < 32; ++i) {
        int idx = i * 32 + t;
        int r = idx >> 6, c = idx & 63;
        int rr = min(r, F_IN - 1);
        float v = W1[(size_t)rr * 64 + c];
        sB[r][c] = (r < F_IN) ? v : 0.0f;
    }
    __syncthreads();

    const int m  = t & 15;
    const int nn = t & 15;
    const int kb = (t < 16) ? 0 : 2;  // per-lane K sub-offset (f32 A/B layout)
    const int hi = (t < 16) ? 0 : 8;  // D row offset per lane half
    const bool full = (row0 + 15) < N;   // wave-uniform

    #pragma unroll
    for (int ct = 0; ct < 4; ++ct) {  // 4 col tiles of 16
        v8f acc = {};
        #pragma unroll
        for (int ks = 0; ks < 4; ++ks) {
            int k0 = ks * 4 + kb;
            v2f a = *(const v2f*)(&sA[m][k0]);       // ds_load_b64
            v2f b;
            b.x = sB[k0][ct * 16 + nn];
            b.y = sB[k0 + 1][ct * 16 + nn];
            acc = __builtin_amdgcn_wmma_f32_16x16x4_f32(false, a, false, b,
                                                        (short)0, acc, false, false);
        }
        // single base address, 8 stores at constant 256B offsets
        float* __restrict__ outp = xl1 + (size_t)(row0 + hi) * 64 + ct * 16 + nn;
        if (full) {
            #pragma unroll
            for (int r = 0; r < 8; ++r) outp[(size_t)r * 64] = acc[r];
        } else {
            #pragma unroll
            for (int r = 0; r < 8; ++r)
                if (row0 + hi + r < N) outp[(size_t)r * 64] = acc[r];
        }
    }
}

// ---- WMMA fp32 GEMM: xl2[N,8] = h1[N,64] @ W2[64,8] (cols padded to 16) ----
__global__ __launch_bounds__(32) void gemm2_wmma(const float* __restrict__ h1,
                                                 const float* __restrict__ W2,
                                                 float* __restrict__ xl2, int N) {
    __shared__ float sB[64][16];  // W2 cols padded 8 -> 16
    const int row0 = blockIdx.x * 16;
    const int t = threadIdx.x;

    #pragma unroll
    for (int i = 0; i < 32; ++i) {
        int idx = i * 32 + t;     // 1024 elems
        int r = idx >> 4, c = idx & 15;
        int cc = min(c, 7);
        float v = W2[(size_t)r * 8 + cc];
        sB[r][c] = (c < 8) ? v : 0.0f;
    }
    __syncthreads();

    const int m  = t & 15;
    const int nn = t & 15;
    const int kb = (t < 16) ? 0 : 2;
    const int hi = (t < 16) ? 0 : 8;
    const int row = min(row0 + m, N - 1);          // clamp: branch-free loads
    const float* __restrict__ arow = h1 + (size_t)row * 64;

    v8f acc = {};
    #pragma unroll
    for (int ks = 0; ks < 16; ++ks) {
        int k0 = ks * 4 + kb;
        v2f a = *(const v2f*)(arow + k0);          // global_load_b64
        v2f b;
        b.x = sB[k0][nn];
        b.y = sB[k0 + 1][nn];
        acc = __builtin_amdgcn_wmma_f32_16x16x4_f32(false, a, false, b,
                                                    (short)0, acc, false, false);
    }
    const bool full = (row0 + 15) < N;             // wave-uniform
    if (nn < 8) {
        float* __restrict__ outp = xl2 + (size_t)(row0 + hi) * 8 + nn;
        if (full) {
            #pragma unroll
            for (int r = 0; r < 8; ++r) outp[(size_t)r * 8] = acc[r];
        } else {
            #pragma unroll
            for (int r = 0; r < 8; ++r)
                if (row0 + hi + r < N) outp[(size_t)r * 8] = acc[r];
        }
    }
}

// ---- per-node attention coefficients, layer 1 ----
__global__ void attn_node1(const float* __restrict__ xl1, const float* __restrict__ as,
                           const float* __restrict__ ad, float* __restrict__ asrc,
                           float* __restrict__ adst, int N) {
    int n = blockIdx.x * blockDim.x + threadIdx.x;
    if (n >= N) return;
    const float* row = xl1 + (size_t)n * 64;
    for (int h = 0; h < 8; ++h) {
        float s = 0.f, d = 0.f;
        #pragma unroll
        for (int c = 0; c < 8; ++c) {
            float v = row[h * 8 + c];
            s += v * as[h * 8 + c];
            d += v * ad[h * 8 + c];
        }
        asrc[n * 8 + h] = s;
        adst[n * 8 + h] = d;
    }
}

// ---- per-node attention coefficients, layer 2 (H2=1) ----
__global__ void attn_node2(const float* __restrict__ xl2, const float* __restrict__ as,
                           const float* __restrict__ ad, float* __restrict__ asrc,
                           float* __restrict__ adst, int N) {
    int n = blockIdx.x * blockDim.x + threadIdx.x;
    if (n >= N) return;
    const float* row = xl2 + (size_t)n * 8;
    float s = 0.f, d = 0.f;
    #pragma unroll
    for (int c = 0; c < 8; ++c) { s += row[c] * as[c]; d += row[c] * ad[c]; }
    asrc[n] = s;
    adst[n] = d;
}

static __device__ __forceinline__ void fetch_edge(int e, const int* __restrict__ ei,
                                                  const float* __restrict__ ea,
                                                  const float* __restrict__ loopA, int E,
                                                  int& s, int& d, float& a0, float& a1) {
    if (e < E) { s = ei[e]; d = ei[E + e]; a0 = ea[2 * e]; a1 = ea[2 * e + 1]; }
    else { int n = e - E; s = n; d = n; a0 = loopA[2 * n]; a1 = loopA[2 * n + 1]; }
}

// ---- layer-1 edge passes: per (edge, head) thread ----
__global__ void edge_max1(const int* ei, const float* ea, const float* loopA,
                          const float* asrc, const float* adst, const float* ve1,
                          float* m1, int E, int ET) {
    int t = blockIdx.x * blockDim.x + threadIdx.x;
    if (t >= ET * 8) return;
    int e = t >> 3, h = t & 7;
    int s, d; float a0, a1;
    fetch_edge(e, ei, ea, loopA, E, s, d, a0, a1);
    float al = asrc[s * 8 + h] + adst[d * 8 + h] + a0 * ve1[h] + a1 * ve1[8 + h];
    al = (al > 0.f) ? al : 0.2f * al;
    atomicMaxF(&m1[d * 8 + h], al);
}
__global__ void edge_sum1(const int* ei, const float* ea, const float* loopA,
                          const float* asrc, const float* adst, const float* ve1,
                          const float* m1, float* z1, int E, int ET) {
    int t = blockIdx.x * blockDim.x + threadIdx.x;
    if (t >= ET * 8) return;
    int e = t >> 3, h = t & 7;
    int s, d; float a0, a1;
    fetch_edge(e, ei, ea, loopA, E, s, d, a0, a1);
    float al = asrc[s * 8 + h] + adst[d * 8 + h] + a0 * ve1[h] + a1 * ve1[8 + h];
    al = (al > 0.f) ? al : 0.2f * al;
    atomicAdd(&z1[d * 8 + h], expf(al - m1[d * 8 + h]));
}
__global__ void edge_agg1(const int* ei, const float* ea, const float* loopA,
                          const float* asrc, const float* adst, const float* ve1,
                          const float* m1, const float* z1, const float* xl1,
                          float* acc1, int E, int ET) {
    int t = blockIdx.x * blockDim.x + threadIdx.x;
    if (t >= ET * 8) return;
    int e = t >> 3, h = t & 7;
    int s, d; float a0, a1;
    fetch_edge(e, ei, ea, loopA, E, s, d, a0, a1);
    float al = asrc[s * 8 + h] + adst[d * 8 + h] + a0 * ve1[h] + a1 * ve1[8 + h];
    al = (al > 0.f) ? al : 0.2f * al;
    float att = expf(al - m1[d * 8 + h]) / (z1[d * 8 + h] + 1e-16f);
    const float* xs = xl1 + (size_t)s * 64 + h * 8;
    float* ap = acc1 + (size_t)d * 64 + h * 8;
    #pragma unroll
    for (int c = 0; c < 8; ++c) atomicAdd(&ap[c], att * xs[c]);
}

// ---- h1 = elu(acc1 + b1) in place ----
__global__ void node_fin1(float* __restrict__ acc1, const float* __restrict__ b1, int N) {
    int t = blockIdx.x * blockDim.x + threadIdx.x;
    if (t >= N * 64) return;
    int j = t & 63;
    float v = acc1[t] + b1[j];
    acc1[t] = (v > 0.f) ? v : expm1f(v);
}

// ---- layer-2 edge passes ----
__global__ void edge_max2(const int* ei, const float* ea, const float* loopA,
                          const float* asrc, const float* adst, const float* ve2,
                          float* m2, int E, int ET) {
    int e = blockIdx.x * blockDim.x + threadIdx.x;
    if (e >= ET) return;
    int s, d; float a0, a1;
    fetch_edge(e, ei, ea, loopA, E, s, d, a0, a1);
    float al = asrc[s] + adst[d] + a0 * ve2[0] + a1 * ve2[1];
    al = (al > 0.f) ? al : 0.2f * al;
    atomicMaxF(&m2[d], al);
}
__global__ void edge_sum2(const int* ei, const float* ea, const float* loopA,
                          const float* asrc, const float* adst, const float* ve2,
                          const float* m2, float* z2, int E, int ET) {
    int e = blockIdx.x * blockDim.x + threadIdx.x;
    if (e >= ET) return;
    int s, d; float a0, a1;
    fetch_edge(e, ei, ea, loopA, E, s, d, a0, a1);
    float al = asrc[s] + adst[d] + a0 * ve2[0] + a1 * ve2[1];
    al = (al > 0.f) ? al : 0.2f * al;
    atomicAdd(&z2[d], expf(al - m2[d]));
}
__global__ void edge_agg2(const int* ei, const float* ea, const float* loopA,
                          const float* asrc, const float* adst, const float* ve2,
                          const float* m2, const float* z2, const float* xl2,
                          float* acc2, int E, int ET) {
    int t = blockIdx.x * blockDim.x + threadIdx.x;
    if (t >= ET * 8) return;
    int e = t >> 3, c = t & 7;
    int s, d; float a0, a1;
    fetch_edge(e, ei, ea, loopA, E, s, d, a0, a1);
    float al = asrc[s] + adst[d] + a0 * ve2[0] + a1 * ve2[1];
    al = (al > 0.f) ? al : 0.2f * al;
    float att = expf(al - m2[d]) / (z2[d] + 1e-16f);
    atomicAdd(&acc2[(size_t)d * 8 + c], att * xl2[(size_t)s * 8 + c]);
}

// ---- h2 = acc2 + b2 (mean over 1 head == identity); gate raw + segment max ----
__global__ void node_fin2(float* __restrict__ acc2, const float* __restrict__ b2,
                          const float* __restrict__ wg, const float* __restrict__ bg,
                          const int* __restrict__ batch, float* __restrict__ graw,
                          float* __restrict__ mg, int N) {
    int n = blockIdx.x * blockDim.x + threadIdx.x;
    if (n >= N) return;
    float g = bg[0];
    #pragma unroll
    for (int c = 0; c < 8; ++c) {
        float v = acc2[(size_t)n * 8 + c] + b2[c];
        acc2[(size_t)n * 8 + c] = v;
        g += v * wg[c];
    }
    graw[n] = g;
    atomicMaxF(&mg[batch[n]], g);
}
__global__ void gate_sum(const float* graw, const int* batch, const float* mg,
                         float* zg, int N) {
    int n = blockIdx.x * blockDim.x + threadIdx.x;
    if (n >= N) return;
    atomicAdd(&zg[batch[n]], expf(graw[n] - mg[batch[n]]));
}
__global__ void gate_fin(const float* graw, const int* batch, const float* mg,
                         const float* zg, const float* acc2, float* pooled,
                         float* out_gate, int N) {
    int n = blockIdx.x * blockDim.x + threadIdx.x;
    if (n >= N) return;
    int b = batch[n];
    float gate = expf(graw[n] - mg[b]) / (zg[b] + 1e-16f);
    out_gate[n] = gate;
    #pragma unroll
    for (int c = 0; c < 8; ++c)
        atomicAdd(&pooled[b * 8 + c], gate * acc2[(size_t)n * 8 + c]);
}
__global__ void cls_kernel(const float* pooled, const float* wc, const float* bc,
                           float* out) {
    int g = blockIdx.x * blockDim.x + threadIdx.x;
    if (g >= GC) return;
    float s = bc[0];
    #pragma unroll
    for (int c = 0; c < 8; ++c) s += pooled[g * 8 + c] * wc[c];
    out[g] = 1.0f / (1.0f + expf(-s));
}

static inline int cdiv(long long a, long long b) { return (int)((a + b - 1) / b); }

extern "C" void kernel_launch(void* const* d_in, const int* in_sizes, int n_in,
                              void* d_out, int out_size, void* d_ws, size_t ws_size,
                              hipStream_t stream) {
    const float* x   = (const float*)d_in[0];
    const int*   ei  = (const int*)  d_in[1];
    const float* ea  = (const float*)d_in[2];
    const int* batch = (const int*)  d_in[3];
    const float* W1  = (const float*)d_in[4];
    const float* as1 = (const float*)d_in[5];
    const float* ad1 = (const float*)d_in[6];
    const float* We1 = (const float*)d_in[7];
    const float* ae1 = (const float*)d_in[8];
    const float* b1  = (const float*)d_in[9];
    const float* W2  = (const float*)d_in[10];
    const float* as2 = (const float*)d_in[11];
    const float* ad2 = (const float*)d_in[12];
    const float* We2 = (const float*)d_in[13];
    const float* ae2 = (const float*)d_in[14];
    const float* b2  = (const float*)d_in[15];
    const float* wg  = (const float*)d_in[16];
    const float* bg  = (const float*)d_in[17];
    const float* wc  = (const float*)d_in[18];
    const float* bc  = (const float*)d_in[19];

    const int N  = in_sizes[0] / F_IN;
    const int E  = in_sizes[1] / 2;
    const int ET = E + N;

    float* ws = (float*)d_ws;
    size_t o = 0;
    float* loopA = ws + o; o += (size_t)N * 2;
    float* loopC = ws + o; o += (size_t)N;
    float* xl1   = ws + o; o += (size_t)N * 64;
    float* asrc1 = ws + o; o += (size_t)N * 8;
    float* adst1 = ws + o; o += (size_t)N * 8;
    float* m1    = ws + o; o += (size_t)N * 8;
    float* z1    = ws + o; o += (size_t)N * 8;
    float* acc1  = ws + o; o += (size_t)N * 64;   // becomes h1
    float* xl2   = ws + o; o += (size_t)N * 8;
    float* asrc2 = ws + o; o += (size_t)N;
    float* adst2 = ws + o; o += (size_t)N;
    float* m2    = ws + o; o += (size_t)N;
    float* z2    = ws + o; o += (size_t)N;
    float* acc2  = ws + o; o += (size_t)N * 8;    // becomes h2
    float* graw  = ws + o; o += (size_t)N;
    float* mg    = ws + o; o += GC;
    float* zg    = ws + o; o += GC;
    float* pooled= ws + o; o += GC * 8;
    float* ve1   = ws + o; o += 16;
    float* ve2   = ws + o; o += 2;

    const int B = 256;
    float* outG    = (float*)d_out;        // [64]
    float* outGate = (float*)d_out + GC;   // [N]

    // ---- init workspace ----
    fill_f32<<<cdiv((size_t)N * 3, B), B, 0, stream>>>(loopA, 0.f, N * 3);      // loopA+loopC contiguous
    fill_f32<<<cdiv((size_t)N * 8, B), B, 0, stream>>>(z1, 0.f, N * 8);
    fill_f32<<<cdiv((size_t)N * 8, B), B, 0, stream>>>(m1, -INFINITY, N * 8);
    fill_f32<<<cdiv((size_t)N * 64, B), B, 0, stream>>>(acc1, 0.f, N * 64);
    fill_f32<<<cdiv(N, B), B, 0, stream>>>(m2, -INFINITY, N);
    fill_f32<<<cdiv(N, B), B, 0, stream>>>(z2, 0.f, N);
    fill_f32<<<cdiv((size_t)N * 8, B), B, 0, stream>>>(acc2, 0.f, N * 8);
    fill_f32<<<1, 64, 0, stream>>>(mg, -INFINITY, GC);
    fill_f32<<<cdiv(GC * 9, B), B, 0, stream>>>(zg, 0.f, GC * 9);               // zg + pooled contiguous

    // ---- precompute ----
    vedge_kernel<<<1, 32, 0, stream>>>(We1, ae1, We2, ae2, ve1, ve2);
    loop_sum_kernel<<<cdiv(E, B), B, 0, stream>>>(ei, ea, loopA, loopC, E);
    loop_div_kernel<<<cdiv(N, B), B, 0, stream>>>(loopA, loopC, N);

    // ---- layer 1 ----
    gemm1_wmma<<<cdiv(N, 16), 32, 0, stream>>>(x, W1, xl1, N);
    attn_node1<<<cdiv(N, B), B, 0, stream>>>(xl1, as1, ad1, asrc1, adst1, N);
    edge_max1<<<cdiv((size_t)ET * 8, B), B, 0, stream>>>(ei, ea, loopA, asrc1, adst1, ve1, m1, E, ET);
    edge_sum1<<<cdiv((size_t)ET * 8, B), B, 0, stream>>>(ei, ea, loopA, asrc1, adst1, ve1, m1, z1, E, ET);
    edge_agg1<<<cdiv((size_t)ET * 8, B), B, 0, stream>>>(ei, ea, loopA, asrc1, adst1, ve1, m1, z1, xl1, acc1, E, ET);
    node_fin1<<<cdiv((size_t)N * 64, B), B, 0, stream>>>(acc1, b1, N);

    // ---- layer 2 ----
    gemm2_wmma<<<cdiv(N, 16), 32, 0, stream>>>(acc1, W2, xl2, N);
    attn_node2<<<cdiv(N, B), B, 0, stream>>>(xl2, as2, ad2, asrc2, adst2, N);
    edge_max2<<<cdiv(ET, B), B, 0, stream>>>(ei, ea, loopA, asrc2, adst2, ve2, m2, E, ET);
    edge_sum2<<<cdiv(ET, B), B, 0, stream>>>(ei, ea, loopA, asrc2, adst2, ve2, m2, z2, E, ET);
    edge_agg2<<<cdiv((size_t)ET * 8, B), B, 0, stream>>>(ei, ea, loopA, asrc2, adst2, ve2, m2, z2, xl2, acc2, E, ET);

    // ---- pooling + classifier ----
    node_fin2<<<cdiv(N, B), B, 0, stream>>>(acc2, b2, wg, bg, batch, graw, mg, N);
    gate_sum<<<cdiv(N, B), B, 0, stream>>>(graw, batch, mg, zg, N);
    gate_fin<<<cdiv(N, B), B, 0, stream>>>(graw, batch, mg, zg, acc2, pooled, outGate, N);
    cls_kernel<<<1, GC, 0, stream>>>(pooled, wc, bc, outG);
}